// RoIPooling2D_50568944943285
// MI455X (gfx1250) — compile-verified
//
#include <hip/hip_runtime.h>

#define ROI_OUT_H 7
#define ROI_OUT_W 7
#define ROI_BINS  (ROI_OUT_H * ROI_OUT_W)
#define XC 256
#define XH 50
#define XW 50
#define CH_CHUNK 4
#define NCHUNK (XC / CH_CHUNK)
#define TILE_ELEMS (CH_CHUNK * XH * XW)   // 10000 floats = 40000 B LDS
#define ROI_SCALE 0.0625f

typedef __attribute__((address_space(1))) int  g_i32;
typedef __attribute__((address_space(3))) int  l_i32;
typedef unsigned int u32x4 __attribute__((ext_vector_type(4)));
typedef int          i32x8 __attribute__((ext_vector_type(8)));
typedef int          i32x4 __attribute__((ext_vector_type(4)));

// ---- fallback: CDNA5 async global->LDS scalar copies ----
__device__ __forceinline__ void async_g2lds_b32(const float* gsrc, float* lds_dst) {
#if __has_builtin(__builtin_amdgcn_global_load_async_to_lds_b32)
    __builtin_amdgcn_global_load_async_to_lds_b32(
        (g_i32*)(unsigned long long)gsrc, (l_i32*)lds_dst, 0, 0);
#else
    unsigned lds_off =
        (unsigned)(unsigned long long)(__attribute__((address_space(3))) float*)lds_dst;
    asm volatile("global_load_async_to_lds_b32 %0, %1, off"
                 :: "v"(lds_off), "v"((unsigned long long)gsrc) : "memory");
#endif
}

__device__ __forceinline__ void wait_asynccnt0() {
#if __has_builtin(__builtin_amdgcn_s_wait_asynccnt)
    __builtin_amdgcn_s_wait_asynccnt(0);
#else
    asm volatile("s_wait_asynccnt 0" ::: "memory");
#endif
}

__global__ __launch_bounds__(256) void roipool_tdm_kernel(
    const float* __restrict__ x,      // (B, 256, 50, 50)
    const float* __restrict__ rois,   // (R, 5): b, x1, y1, x2, y2
    float* __restrict__ out)          // (R, 256, 7, 7)
{
    __shared__ float tile[TILE_ELEMS];

    const int blk   = blockIdx.x;
    const int r     = blk / NCHUNK;
    const int chunk = blk - r * NCHUNK;
    const int c0    = chunk * CH_CHUNK;
    const int tid   = threadIdx.x;

    const float* roi = rois + r * 5;
    const int b   = (int)roi[0];
    const int rx1 = (int)(roi[1] * ROI_SCALE);   // trunc == floor (non-neg data)
    const int ry1 = (int)(roi[2] * ROI_SCALE);
    int rx2 = (int)(roi[3] * ROI_SCALE);
    int ry2 = (int)(roi[4] * ROI_SCALE);
    rx2 = min(max(rx2, 0), XW - 1);
    ry2 = min(max(ry2, 0), XH - 1);
    const int h = ry2 - ry1 + 1;
    const int w = rx2 - rx1 + 1;

    const int y_lo = max(ry1, 0), y_hi = ry2 + 1;
    const int x_lo = max(rx1, 0), x_hi = rx2 + 1;
    const int h_reg = y_hi - y_lo;
    const int w_reg = x_hi - x_lo;

    float* outp = out + ((long long)r * XC + c0) * ROI_BINS;

    if (h_reg <= 0 || w_reg <= 0) {   // empty mask -> -inf everywhere
        for (int idx = tid; idx < CH_CHUNK * ROI_BINS; idx += 256)
            outp[idx] = -__builtin_inff();
        return;
    }

    const float* xb = x + (((long long)b * XC + c0) * XH) * XW;

#if __has_builtin(__builtin_amdgcn_tensor_load_to_lds)
    // ---- TDM: one DMA descriptor moves the whole (4, h_reg, w_reg) tile ----
    if (tid < 32) {   // wave 0 only; TENSORcnt is per-wave, EXEC ignored by TDM
        const unsigned long long gaddr =
            (unsigned long long)(xb + (long long)y_lo * XW + x_lo);
        const unsigned lds_base =
            (unsigned)(unsigned long long)(__attribute__((address_space(3))) float*)&tile[0];

        // D# group 0: count=1 | lds_addr | global_addr[56:0] | type=2 ("image")
        u32x4 g0 = { 1u,
                     lds_base,
                     (unsigned)(gaddr & 0xFFFFFFFFu),
                     (unsigned)((gaddr >> 32) & 0x01FFFFFFu) | (2u << 30) };

        // D# group 1: data_size=4B; tensor_dim0/1 = W/H; tile_dim0/1/2 = w_reg/h_reg/4;
        // tensor_dim0_stride = W; tensor_dim1_stride = H*W
        const unsigned s0 = XW;         // dim0 stride (elements), fits in 32 bits
        const unsigned s1 = XH * XW;    // dim1 stride (elements)
        i32x8 g1 = {
            (int)(2u << 16),                                               // data_size=2 (4B)
            (int)(((unsigned)XW & 0xFFFFu) << 16),                         // tdim0[15:0]
            (int)((((unsigned)XW >> 16) & 0xFFFFu) | (((unsigned)XH & 0xFFFFu) << 16)),
            (int)((((unsigned)XH >> 16) & 0xFFFFu) | ((unsigned)w_reg << 16)),  // tile_dim0
            (int)((unsigned)h_reg | ((unsigned)CH_CHUNK << 16)),           // tile_dim1|tile_dim2
            (int)s0,                                                       // stride0[31:0]
            (int)((s1 & 0xFFFFu) << 16),                                   // stride0[47:32]=0 | stride1[15:0]
            (int)(s1 >> 16)                                                // stride1[47:16]
        };

        // D# group 2: tensor_dim2 = 256 channels; tensor_dim2_stride = H*W; tile_dim3=0
        i32x4 g2 = { XC, 0, XH * XW, 0 };
        i32x4 g3 = { 0, 0, 0, 0 };
        i32x8 g4 = { 0, 0, 0, 0, 0, 0, 0, 0 };   // unused extended group (VADDR4: "set to NULL")

        __builtin_amdgcn_tensor_load_to_lds(g0, g1, g2, g3, g4, /*cpol=*/0);
        __builtin_amdgcn_s_wait_tensorcnt(0);
    }
    __syncthreads();
#else
    // ---- fallback: cooperative async element copies ----
    const int plane = h_reg * w_reg;
    const int nelem = CH_CHUNK * plane;
    for (int e = tid; e < nelem; e += 256) {
        const int cl  = e / plane;
        const int rem = e - cl * plane;
        const int yy  = rem / w_reg;
        const int xx  = rem - yy * w_reg;
        const float* g = xb + ((long long)cl * XH + (y_lo + yy)) * XW + (x_lo + xx);
        async_g2lds_b32(g, &tile[e]);
    }
    wait_asynccnt0();
    __syncthreads();
#endif

    // ---- 196 outputs: one (channel, bin) max each, coalesced stores ----
    for (int idx = tid; idx < CH_CHUNK * ROI_BINS; idx += 256) {
        const int cl  = idx / ROI_BINS;
        const int bin = idx - cl * ROI_BINS;
        const int i   = bin / ROI_OUT_W;
        const int j   = bin - i * ROI_OUT_W;
        // ys = y1 + i*h//7 ; ye = y1 + ceil((i+1)*h/7)   (h >= 1 here)
        const int ys = max(ry1 + (i * h) / ROI_OUT_H, y_lo);
        const int ye = min(ry1 + ((i + 1) * h + ROI_OUT_H - 1) / ROI_OUT_H, y_hi);
        const int xs = max(rx1 + (j * w) / ROI_OUT_W, x_lo);
        const int xe = min(rx1 + ((j + 1) * w + ROI_OUT_W - 1) / ROI_OUT_W, x_hi);

        float m = -__builtin_inff();
        for (int y = ys; y < ye; ++y) {
            const float* trow = &tile[(cl * h_reg + (y - y_lo)) * w_reg - x_lo];
            for (int xq = xs; xq < xe; ++xq)
                m = fmaxf(m, trow[xq]);
        }
        outp[idx] = m;
    }
}

extern "C" void kernel_launch(void* const* d_in, const int* in_sizes, int n_in,
                              void* d_out, int out_size, void* d_ws, size_t ws_size,
                              hipStream_t stream) {
    const float* x    = (const float*)d_in[0];
    const float* rois = (const float*)d_in[1];
    float* out        = (float*)d_out;
    const int R = in_sizes[1] / 5;           // 256 ROIs
    dim3 grid(R * NCHUNK);                    // 256 * 64 = 16384 blocks
    roipool_tdm_kernel<<<grid, 256, 0, stream>>>(x, rois, out);
}